// SpatialToMe_30434138260169
// MI455X (gfx1250) — compile-verified
//
#include <hip/hip_runtime.h>

// Problem constants (from reference: B,N,D,L = 8,8192,512,512; REDUCE_RATIO=0.3)
#define B_ 8
#define N_ 8192
#define D_ 512
#define L_ 512
#define NSRC_ 4096           // N/2
#define KSEL_ 2457           // int(N*0.3)
#define NNEW_ 5735           // N - KSEL

typedef float v2f __attribute__((ext_vector_type(2)));
typedef float v8f __attribute__((ext_vector_type(8)));

// ---------------------------------------------------------------------------
// Kernel 1: metric[b,n] = sum_d x_feat[b,n,d]  via V_WMMA_F32_16X16X4_F32.
// A = 16x4 f32 tile of rows, B = ones(4x16)  =>  D[m, *] = row sum of m.
// Per ISA layout: A lanes 0-15 hold M=0..15 {K=0 in v0, K=1 in v1},
//                 lanes 16-31 hold M=0..15 {K=2, K=3}.
// C/D layout: VGPR j: lanes 0-15 -> M=j, lanes 16-31 -> M=j+8.
// (Mean's 1/D and the per-batch /norm are positive per-batch scalars: the
// downstream argmax / top_k selection is invariant to them, so we skip both.)
// ---------------------------------------------------------------------------
__global__ __launch_bounds__(256) void metric_kernel(const float* __restrict__ xf,
                                                     float* __restrict__ metric) {
  const int wave = threadIdx.x >> 5;
  const int lane = threadIdx.x & 31;
  const long rowBase = (long)blockIdx.x * 128 + (long)wave * 16;  // row = b*N + n
  const int r  = lane & 15;
  const int kh = (lane >> 4) * 2;                 // 0 for lanes 0-15, 2 for 16-31
  const float* p = xf + (rowBase + r) * (long)D_ + kh;

  v8f c = {0.f,0.f,0.f,0.f,0.f,0.f,0.f,0.f};
  v2f bm; bm.x = 1.0f; bm.y = 1.0f;               // ones: B layout irrelevant
  #pragma unroll 4
  for (int k = 0; k < D_; k += 4) {
    v2f a; a.x = p[k]; a.y = p[k + 1];
    c = __builtin_amdgcn_wmma_f32_16x16x4_f32(false, a, false, bm,
                                              (short)0, c, false, false);
  }
  if (lane == 0 || lane == 16) {
    const int off = (lane == 16) ? 8 : 0;
    #pragma unroll
    for (int j = 0; j < 8; j++) metric[rowBase + off + j] = c[j];
  }
}

// ---------------------------------------------------------------------------
// Block-wide inclusive scan over `len` ints (len = 4096 or 8192), 1024 threads.
// ---------------------------------------------------------------------------
__device__ __forceinline__ void scan_inclusive(int* data, int len, int* part) {
  const int tid = threadIdx.x;
  const int ept = len >> 10;                      // elems per thread (4 or 8)
  const int base = tid * ept;
  int sum = 0;
  #pragma unroll
  for (int i = 0; i < 8; i++)
    if (i < ept) { sum += data[base + i]; data[base + i] = sum; }
  part[tid] = sum;
  __syncthreads();
  for (int off = 1; off < 1024; off <<= 1) {      // Hillis-Steele over partials
    int v  = part[tid];
    int vo = (tid >= off) ? part[tid - off] : 0;
    __syncthreads();
    part[tid] = v + vo;
    __syncthreads();
  }
  const int excl = (tid > 0) ? part[tid - 1] : 0;
  #pragma unroll
  for (int i = 0; i < 8; i++)
    if (i < ept) data[base + i] += excl;
  __syncthreads();
}

// ---------------------------------------------------------------------------
// Kernel 2: per-batch selection. One 1024-thread block per batch.
// scores = rank-1 outer product => only need max/argmax/min/argmin of b.
// top_k set => radix-select k-th largest order-key + lowest-index tie quota.
// node_idx takes at most 3 distinct values per batch (argmax_b, argmin_b, 0),
// so segment counts need only 3 shared counters.
// ---------------------------------------------------------------------------
__global__ __launch_bounds__(1024) void select_kernel(
    const float* __restrict__ metric, int* __restrict__ counts,
    int* __restrict__ keep_t, int* __restrict__ msrc, int* __restrict__ mdst,
    float* __restrict__ unpool_out) {
  __shared__ unsigned int keys[NSRC_];   // 16 KB
  __shared__ int carr[N_];               // 32 KB (reduction scratch / scans)
  __shared__ int part[1024];             // 4 KB
  __shared__ int hist[256];              // 1 KB
  __shared__ int sh_prefix, sh_rem, sh_amax, sh_amin, sh_mc, sh_cp, sh_cn, sh_cz;
  __shared__ float sh_bmax, sh_bmin;

  const int b = blockIdx.x, tid = threadIdx.x;
  const float* m = metric + (long)b * N_;

  // --- 1. extrema of b = metric at odd positions (first-index tie-break) ---
  float lmaxv = -INFINITY, lminv = INFINITY; int lmaxi = 0, lmini = 0;
  #pragma unroll
  for (int i = 0; i < 4; i++) {
    int s = tid * 4 + i; float v = m[2 * s + 1];
    if (v > lmaxv) { lmaxv = v; lmaxi = s; }
    if (v < lminv) { lminv = v; lmini = s; }
  }
  carr[tid]        = __float_as_int(lmaxv); carr[1024 + tid] = lmaxi;
  carr[2048 + tid] = __float_as_int(lminv); carr[3072 + tid] = lmini;
  __syncthreads();
  for (int off = 512; off > 0; off >>= 1) {
    if (tid < off) {
      if (__int_as_float(carr[tid + off]) > __int_as_float(carr[tid])) {
        carr[tid] = carr[tid + off]; carr[1024 + tid] = carr[1024 + tid + off];
      }
      if (__int_as_float(carr[2048 + tid + off]) < __int_as_float(carr[2048 + tid])) {
        carr[2048 + tid] = carr[2048 + tid + off]; carr[3072 + tid] = carr[3072 + tid + off];
      }
    }
    __syncthreads();
  }
  if (tid == 0) {
    sh_bmax = __int_as_float(carr[0]);    sh_amax = carr[1024];
    sh_bmin = __int_as_float(carr[2048]); sh_amin = carr[3072];
    sh_mc = 0; sh_cp = 0; sh_cn = 0; sh_cz = 0;
  }
  __syncthreads();
  const float bmax = sh_bmax, bmin = sh_bmin;
  const int amax = sh_amax, amin = sh_amin;

  // --- 2. order keys of node_max (monotonic float->uint map) ---
  #pragma unroll
  for (int i = 0; i < 4; i++) {
    int s = tid * 4 + i; float a = m[2 * s];
    float nm = (a > 0.f) ? a * bmax : ((a < 0.f) ? a * bmin : 0.f);
    unsigned ub = __float_as_uint(nm);
    keys[s] = (ub & 0x80000000u) ? ~ub : (ub | 0x80000000u);
  }
  __syncthreads();

  // --- 3. radix-select k-th largest key (MSB-first, 8-bit digits) ---
  unsigned prefix = 0; int rem = KSEL_;
  for (int p = 3; p >= 0; p--) {
    if (tid < 256) hist[tid] = 0;
    __syncthreads();
    #pragma unroll
    for (int i = 0; i < 4; i++) {
      unsigned kk = keys[tid * 4 + i];
      bool match = (p == 3) || ((kk >> ((p + 1) * 8)) == prefix);
      if (match) atomicAdd(&hist[(kk >> (p * 8)) & 255], 1);
    }
    __syncthreads();
    if (tid == 0) {
      int g = 0, bsel = 0;
      for (int bb = 255; bb >= 0; bb--) {
        if (g + hist[bb] >= rem) { bsel = bb; break; }
        g += hist[bb];
      }
      sh_prefix = (int)((prefix << 8) | (unsigned)bsel);
      sh_rem = rem - g;
    }
    __syncthreads();
    prefix = (unsigned)sh_prefix; rem = sh_rem;
    __syncthreads();
  }
  const unsigned T = prefix; const int quota = rem;  // take `quota` keys == T

  // --- 4. rank among equal keys (lowest index first) ---
  #pragma unroll
  for (int i = 0; i < 4; i++) { int s = tid * 4 + i; carr[s] = (keys[s] == T) ? 1 : 0; }
  __syncthreads();
  scan_inclusive(carr, NSRC_, part);

  // --- 5. selected mask, overwritten into keys[] ---
  #pragma unroll
  for (int i = 0; i < 4; i++) {
    int s = tid * 4 + i; unsigned kk = keys[s];
    int sel = (kk > T) || (kk == T && (carr[s] - 1) < quota);
    keys[s] = (unsigned)sel;
  }
  __syncthreads();

  // --- 6. presence -> inclusive cumsum (compact = carr - 1) ---
  #pragma unroll
  for (int i = 0; i < 8; i++) {
    int n = tid * 8 + i;
    carr[n] = (n & 1) ? 1 : (keys[n >> 1] ? 0 : 1);
  }
  __syncthreads();
  scan_inclusive(carr, N_, part);

  // --- 7a. merged list + 3-way class counts ---
  #pragma unroll
  for (int i = 0; i < 4; i++) {
    int s = tid * 4 + i;
    if (keys[s]) {
      float a = m[2 * s];
      int ni = (a > 0.f) ? amax : ((a < 0.f) ? amin : 0);
      int t  = carr[2 * ni + 1] - 1;
      int idx = atomicAdd(&sh_mc, 1);
      msrc[b * KSEL_ + idx] = 2 * s;
      mdst[b * KSEL_ + idx] = t;
      if (a > 0.f) atomicAdd(&sh_cp, 1);
      else if (a < 0.f) atomicAdd(&sh_cn, 1);
      else atomicAdd(&sh_cz, 1);
    }
  }
  __syncthreads();
  const int tp = carr[2 * amax + 1] - 1;
  const int tn = carr[2 * amin + 1] - 1;
  const int tz = carr[1] - 1;
  const int cp = sh_cp, cn = sh_cn, cz = sh_cz;

  // --- 7b. unpool / keep_t / counts (each written exactly once) ---
  #pragma unroll
  for (int i = 0; i < 8; i++) {
    int n = tid * 8 + i;
    int gid;
    if (n & 1) gid = n;
    else if (keys[n >> 1]) {
      float a = m[n];  // n even => m[2*(n>>1)]
      int ni = (a > 0.f) ? amax : ((a < 0.f) ? amin : 0);
      gid = 2 * ni + 1;
    } else gid = n;
    int up = carr[gid] - 1;
    unpool_out[(long)b * N_ + n] = (float)up;
    int kt = (gid == n) ? up : -1;
    keep_t[(long)b * N_ + n] = kt;
    if (kt >= 0) {
      int c = 1;
      if (kt == tp) c += cp;
      if (kt == tn) c += cn;
      if (kt == tz) c += cz;
      counts[(long)b * NNEW_ + kt] = c;
    }
  }
}

// ---------------------------------------------------------------------------
// Kernel 3: copy kept rows (scaled by 1/(count+1e-6)). One block per old row;
// waves 0-3 do the D_ feat floats, waves 4-7 the L_ raw floats (float4).
// ---------------------------------------------------------------------------
__global__ __launch_bounds__(256) void copy_kernel(
    const float* __restrict__ xf, const float* __restrict__ xr,
    const int* __restrict__ keep_t, const int* __restrict__ counts,
    float* __restrict__ out_feat, float* __restrict__ out_raw) {
  const long row = blockIdx.x;                 // b*N + n
  const long b = row >> 13;
  const int t = keep_t[row];
  if (t < 0) return;
  const float scale = 1.0f / ((float)counts[b * NNEW_ + t] + 1e-6f);
  const int tid = threadIdx.x;
  if (tid < 128) {
    const float4* s4 = (const float4*)(xf + row * D_);
    float4* d4 = (float4*)(out_feat + (b * NNEW_ + t) * (long)D_);
    float4 v = s4[tid];
    v.x *= scale; v.y *= scale; v.z *= scale; v.w *= scale;
    d4[tid] = v;
  } else {
    const int j = tid - 128;
    const float4* s4 = (const float4*)(xr + row * L_);
    float4* d4 = (float4*)(out_raw + (b * NNEW_ + t) * (long)L_);
    float4 v = s4[j];
    v.x *= scale; v.y *= scale; v.z *= scale; v.w *= scale;
    d4[j] = v;
  }
}

// ---------------------------------------------------------------------------
// Kernel 4: atomically add merged (scaled) rows. One block per merged row.
// ---------------------------------------------------------------------------
__global__ __launch_bounds__(256) void merge_kernel(
    const float* __restrict__ xf, const float* __restrict__ xr,
    const int* __restrict__ msrc, const int* __restrict__ mdst,
    const int* __restrict__ counts,
    float* __restrict__ out_feat, float* __restrict__ out_raw) {
  const int i = blockIdx.x;
  const int b = i / KSEL_, j = i % KSEL_;
  const int n = msrc[b * KSEL_ + j];
  const int t = mdst[b * KSEL_ + j];
  const float scale = 1.0f / ((float)counts[(long)b * NNEW_ + t] + 1e-6f);
  const long src = (long)b * N_ + n;
  const long dst = (long)b * NNEW_ + t;
  const int tid = threadIdx.x;
  #pragma unroll
  for (int q = 0; q < 2; q++) {
    int c = tid * 2 + q;
    float v = xf[src * D_ + c] * scale;
    __hip_atomic_fetch_add(&out_feat[dst * D_ + c], v,
                           __ATOMIC_RELAXED, __HIP_MEMORY_SCOPE_AGENT);
    float w = xr[src * L_ + c] * scale;
    __hip_atomic_fetch_add(&out_raw[dst * L_ + c], w,
                           __ATOMIC_RELAXED, __HIP_MEMORY_SCOPE_AGENT);
  }
}

// ---------------------------------------------------------------------------
extern "C" void kernel_launch(void* const* d_in, const int* in_sizes, int n_in,
                              void* d_out, int out_size, void* d_ws, size_t ws_size,
                              hipStream_t stream) {
  const float* xf = (const float*)d_in[0];   // x_feat (B,N,D) f32
  const float* xr = (const float*)d_in[1];   // x_raw  (B,N,L) f32

  float* out_feat = (float*)d_out;                                   // B*NNEW*D
  float* out_raw  = out_feat + (size_t)B_ * NNEW_ * D_;              // B*NNEW*L
  float* unpool   = out_raw  + (size_t)B_ * NNEW_ * L_;              // B*N

  // workspace layout (~865 KB)
  float* metric = (float*)d_ws;                                      // B*N
  int* counts = (int*)(metric + (size_t)B_ * N_);                    // B*NNEW
  int* keep_t = counts + (size_t)B_ * NNEW_;                         // B*N
  int* msrc   = keep_t + (size_t)B_ * N_;                            // B*KSEL
  int* mdst   = msrc + (size_t)B_ * KSEL_;                           // B*KSEL

  metric_kernel<<<(B_ * N_) / 128, 256, 0, stream>>>(xf, metric);
  select_kernel<<<B_, 1024, 0, stream>>>(metric, counts, keep_t, msrc, mdst, unpool);
  copy_kernel<<<B_ * N_, 256, 0, stream>>>(xf, xr, keep_t, counts, out_feat, out_raw);
  merge_kernel<<<B_ * KSEL_, 256, 0, stream>>>(xf, xr, msrc, mdst, counts, out_feat, out_raw);
}